// MetaQDA_FB_FixShot_67748814127642
// MI455X (gfx1250) — compile-verified
//
#include <hip/hip_runtime.h>
#include <hip/hip_bf16.h>
#include <math.h>

#define DD 512
#define CC 64
#define SHOTS 5
#define QQ 2048

typedef __attribute__((ext_vector_type(2))) float v2f;
typedef __attribute__((ext_vector_type(8))) float v8f;
typedef unsigned int u32x4 __attribute__((ext_vector_type(4)));
typedef int i32x8 __attribute__((ext_vector_type(8)));
typedef int i32x4 __attribute__((ext_vector_type(4)));

// D = A(16x4) x B(4x16) + C, full fp32 (V_WMMA_F32_16X16X4_F32)
__device__ __forceinline__ v8f wmma_f32_k4(v2f a, v2f b, v8f c) {
    return __builtin_amdgcn_wmma_f32_16x16x4_f32(
        /*neg_a=*/false, a, /*neg_b=*/false, b,
        /*c_mod=*/(short)0, c, /*reuse_a=*/false, /*reuse_b=*/false);
}

// ---------------- stage 1: scalar parameters ----------------
// scal[0]=kappa_n [1]=wm(|k+1e-6|/kn) [2]=wdm((|k|+1e-6)/kn) [3]=sp
// [4]=bias_shared [5]=xmean_weight [6]=sigma_scale(kn*sp/(kn+1))
__global__ void scalars_kernel(const float* __restrict__ kappa,
                               const float* __restrict__ nu,
                               float* __restrict__ scal) {
    float ka = kappa[0];
    float kn = fabsf(ka) + 1e-6f + 5.0f;
    float nuc = fmaxf(nu[0], (float)(DD - 1) + 1e-6f);
    float sp = nuc + 5.0f - (float)DD + 2.0f;
    scal[0] = kn;
    scal[1] = fabsf(ka + 1e-6f) / kn;
    scal[2] = (fabsf(ka) + 1e-6f) / kn;
    scal[3] = sp;
    scal[4] = lgammaf(0.5f * (sp + (float)DD)) - lgammaf(0.5f * sp)
            - 0.5f * (float)DD * logf(sp);
    scal[5] = 5.0f / kn;
    scal[6] = kn * sp / (kn + 1.0f);
}

// ---------------- stage 2: build L = diag(|d|) + strict-lower ----------------
__global__ void build_L_kernel(const float* __restrict__ diag,
                               const float* __restrict__ lower,
                               float* __restrict__ L) {
    int idx = blockIdx.x * blockDim.x + threadIdx.x;  // 512*512 threads
    int i = idx >> 9, j = idx & 511;
    float v = (i == j) ? fabsf(diag[i]) : ((j < i) ? lower[idx] : 0.0f);
    L[idx] = v;
}

// ---------------- stage 3: scatter = L * L^T (fp32 WMMA) ----------------
// one wave per 16x16 output tile; 32x32 = 1024 tiles
__global__ void llt_kernel(const float* __restrict__ L,
                           float* __restrict__ scatter) {
    const int wave = threadIdx.x >> 5;
    const int lane = threadIdx.x & 31;
    const int task = blockIdx.x * (blockDim.x >> 5) + wave;  // 0..1023
    const int ti = task >> 5;        // row tile
    const int tj = task & 31;        // col tile
    const int hl  = lane >> 4;       // half of wave
    const int l16 = lane & 15;

    const int am = ti * 16 + l16;    // A row (per-lane, both halves M=0..15)
    const int bn = tj * 16 + l16;    // B col

    v8f acc = {0.f, 0.f, 0.f, 0.f, 0.f, 0.f, 0.f, 0.f};
    for (int k0 = 0; k0 < DD; k0 += 4) {
        const int ka = k0 + 2 * hl;
        v2f a, b;
        a.x = L[am * DD + ka];
        a.y = L[am * DD + ka + 1];
        // B[k][n] = (L^T)[k][n] = L[n][k]
        b.x = L[bn * DD + ka];
        b.y = L[bn * DD + ka + 1];
        acc = wmma_f32_k4(a, b, acc);
    }
    // C layout: lane holds col n=l16; VGPR r holds row r + 8*half
    for (int r = 0; r < 8; ++r) {
        int m = ti * 16 + r + 8 * hl;
        scatter[m * DD + tj * 16 + l16] = acc[r];
    }
}

// ---------------- stage 4: class means and mu ----------------
__global__ void stats_kernel(const float* __restrict__ Xs,
                             const float* __restrict__ m,
                             const float* __restrict__ scal,
                             float* __restrict__ x_mean,
                             float* __restrict__ mu) {
    int idx = blockIdx.x * blockDim.x + threadIdx.x;   // C*D = 32768
    int c = idx >> 9, d = idx & 511;
    float s = 0.f;
    for (int sh = 0; sh < SHOTS; ++sh) s += Xs[(c * SHOTS + sh) * DD + d];
    float xm = s * (1.0f / SHOTS);
    x_mean[idx] = xm;
    mu[idx] = scal[1] * m[d] + xm * scal[5];
}

// ---------------- stage 5: assemble sigma [C][D][D] ----------------
__global__ void sigma_kernel(const float* __restrict__ Xs,
                             const float* __restrict__ m,
                             const float* __restrict__ scal,
                             const float* __restrict__ scatter,
                             const float* __restrict__ x_mean,
                             float* __restrict__ sigma) {
    size_t idx = (size_t)blockIdx.x * blockDim.x + threadIdx.x; // c*D*D+i*D+j
    int j = (int)(idx & 511);
    size_t t = idx >> 9;
    int i = (int)(t & 511);
    int c = (int)(t >> 9);
    const float* Xc = Xs + (size_t)c * SHOTS * DD;
    float mo = 0.f;
    for (int sh = 0; sh < SHOTS; ++sh)
        mo += Xc[sh * DD + i] * Xc[sh * DD + j];
    mo *= (1.0f / SHOTS);
    float dmi = x_mean[c * DD + i] - m[i];
    float dmj = x_mean[c * DD + j] - m[j];
    sigma[idx] = (scatter[i * DD + j] + mo + scal[2] * dmi * dmj) * scal[6];
}

// ---------------- stage 6: in-place Gauss-Jordan inverse + logdet ----------------
// one workgroup (512 threads) per class; sigma is SPD -> no pivoting needed
__global__ void gj_inverse_kernel(float* __restrict__ sigma,
                                  const float* __restrict__ scal,
                                  float* __restrict__ biases) {
    __shared__ float rowk[DD];
    __shared__ float sh_invd;
    __shared__ float sh_logdet;
    const int c = blockIdx.x;
    float* A = sigma + (size_t)c * DD * DD;
    const int t = threadIdx.x;     // 0..511, owns column t (scale) and row t (elim)
    if (t == 0) sh_logdet = 0.f;

    for (int k = 0; k < DD; ++k) {
        __syncthreads();                        // prior elimination complete
        if (t == 0) {
            float d = A[(size_t)k * DD + k];
            sh_invd = 1.0f / d;
            sh_logdet += logf(fabsf(d));
        }
        __syncthreads();
        const float invd = sh_invd;
        // scale row k (with A[k][k] -> 1 substitution); stage row in LDS
        float v = ((t == k) ? 1.0f : A[(size_t)k * DD + t]) * invd;
        A[(size_t)k * DD + t] = v;
        rowk[t] = v;
        __syncthreads();
        // eliminate: thread t owns row t
        if (t != k) {
            float f = A[(size_t)t * DD + k];
            float* Ar = A + (size_t)t * DD;
            for (int j = 0; j < DD; ++j) {
                float cur = (j == k) ? 0.0f : Ar[j];
                Ar[j] = fmaf(-f, rowk[j], cur);
            }
        }
    }
    __syncthreads();
    if (t == 0) biases[c] = scal[4] - 0.5f * sh_logdet;
}

// ---------------- stage 7: Mahalanobis + head (fp32 WMMA, TDM-fed LDS) ----------
// block = (class c, 64-query group); 4 waves x 16 queries. The 512x16 B panel
// (sigma_inv columns) is DMA'd into LDS once per n-tile by the Tensor Data
// Mover and shared by all 4 waves.
__global__ void predict_kernel(const float* __restrict__ Xq,
                               const float* __restrict__ mu,
                               const float* __restrict__ Sinv,
                               const float* __restrict__ biases,
                               const float* __restrict__ scal,
                               float* __restrict__ out) {
    __shared__ float Blds[DD * 16];            // 32 KB B panel [k][n]
    const int wave = threadIdx.x >> 5;
    const int lane = threadIdx.x & 31;
    const int c  = blockIdx.x >> 5;            // class (64)
    const int qg = blockIdx.x & 31;            // query group (32)
    const int q0 = qg * 64 + wave * 16;        // this wave's 16-query tile
    const int hl  = lane >> 4;
    const int l16 = lane & 15;
    const float sp = scal[3];
    const float* S   = Sinv + (size_t)c * DD * DD;
    const float* muc = mu + c * DD;

    float dist[8];
    for (int r = 0; r < 8; ++r) dist[r] = 0.f;
    const int am = q0 + l16;                   // this lane's A row (query)

    for (int nt = 0; nt < 32; ++nt) {
        const int n0 = nt * 16;
        __syncthreads();                       // previous panel fully consumed
#if __has_builtin(__builtin_amdgcn_tensor_load_to_lds) && __has_builtin(__builtin_amdgcn_s_wait_tensorcnt)
        if (threadIdx.x < 32) {                // wave 0 issues the TDM
            unsigned lds_base = (unsigned)(size_t)(void*)&Blds[0];
            unsigned long long ga =
                (unsigned long long)(size_t)(const void*)(S + n0);
            // D# group0: count=1 | lds_addr | global_addr(57b) | type=2
            u32x4 g0 = {1u, lds_base,
                        (unsigned)(ga & 0xFFFFFFFFu),
                        ((unsigned)((ga >> 32) & 0x01FFFFFFu)) | 0x80000000u};
            // D# group1: data_size=4B, tensor 512x512 stride 512, tile 16x512
            i32x8 g1;
            g1[0] = 0x20000;              // wg_mask=0, data_size=2 (4 bytes)
            g1[1] = (int)(512u << 16);    // tensor_dim0[15:0]=512 in [31:16]
            g1[2] = (int)(512u << 16);    // td0[31:16]=0; tensor_dim1[15:0]=512
            g1[3] = (int)(16u << 16);     // td1[31:16]=0; tile_dim0=16
            g1[4] = 512;                  // tile_dim1=512 (rows); tile_dim2=0
            g1[5] = 512;                  // tensor_dim0_stride[31:0]=512
            g1[6] = 0;                    // stride hi / tensor_dim1_stride lo
            g1[7] = 0;                    // tensor_dim1_stride hi (2D: unused)
            i32x4 z4 = {0, 0, 0, 0};      // groups 2/3 unused for 2D tensors
            i32x8 z8 = {0, 0, 0, 0, 0, 0, 0, 0};
            __builtin_amdgcn_tensor_load_to_lds(g0, g1, z4, z4, z8, 0);
            __builtin_amdgcn_s_wait_tensorcnt(0);
        }
#else
        for (int e = threadIdx.x; e < DD * 16; e += blockDim.x)
            Blds[e] = S[(size_t)(e >> 4) * DD + n0 + (e & 15)];
#endif
        __syncthreads();                       // panel visible to all waves

        v8f acc = {0.f, 0.f, 0.f, 0.f, 0.f, 0.f, 0.f, 0.f};
        for (int k0 = 0; k0 < DD; k0 += 4) {
            const int ka = k0 + 2 * hl;
            v2f a, b;
            a.x = Xq[am * DD + ka]     - muc[ka];
            a.y = Xq[am * DD + ka + 1] - muc[ka + 1];
            b.x = Blds[ka * 16 + l16];         // ds_load-fed B fragment
            b.y = Blds[(ka + 1) * 16 + l16];
            acc = wmma_f32_k4(a, b, acc);
        }
        // fuse: dist[q] += tmp[q][n] * diff[q][n] for this lane's column n
        const int n = n0 + l16;
        const float mun = muc[n];
        for (int r = 0; r < 8; ++r) {
            const int q = q0 + r + 8 * hl;
            dist[r] = fmaf(acc[r], Xq[q * DD + n] - mun, dist[r]);
        }
    }
    // reduce over the 16 lanes of each half (columns n)
    for (int r = 0; r < 8; ++r) {
        float v = dist[r];
        v += __shfl_xor(v, 1, 16);
        v += __shfl_xor(v, 2, 16);
        v += __shfl_xor(v, 4, 16);
        v += __shfl_xor(v, 8, 16);
        dist[r] = v;
    }
    if (l16 == 0) {
        const float bias = biases[c];
        const float coef = 0.5f * (sp + (float)DD);
        for (int r = 0; r < 8; ++r) {
            const int q = q0 + r + 8 * hl;
            out[q * CC + c] = bias - coef * logf(1.0f + dist[r] / sp);
        }
    }
}

extern "C" void kernel_launch(void* const* d_in, const int* in_sizes, int n_in,
                              void* d_out, int out_size, void* d_ws, size_t ws_size,
                              hipStream_t stream) {
    const float* Xs    = (const float*)d_in[0];   // [C*SHOTS, D]
    // d_in[1] = y (labels, contiguous by construction) -- unused
    const float* Xq    = (const float*)d_in[2];   // [Q, D]
    const float* m     = (const float*)d_in[3];   // [1, D]
    const float* kappa = (const float*)d_in[4];   // scalar
    const float* nu    = (const float*)d_in[5];   // scalar
    const float* diag  = (const float*)d_in[6];   // [D]
    const float* lower = (const float*)d_in[7];   // [D, D]
    float* out = (float*)d_out;                   // [Q, C]

    char* ws = (char*)d_ws;
    float* L       = (float*)(ws);                                      // 1 MB
    float* scatter = (float*)(ws + (1ull << 20));                       // 1 MB
    float* x_mean  = (float*)(ws + (2ull << 20));                       // 128 KB
    float* mu      = (float*)(ws + (2ull << 20) + (128ull << 10));      // 128 KB
    float* biases  = (float*)(ws + (2ull << 20) + (256ull << 10));      // 256 B
    float* scal    = (float*)(ws + (2ull << 20) + (256ull << 10) + 1024);
    float* sigma   = (float*)(ws + (4ull << 20));                       // 64 MB

    scalars_kernel<<<1, 1, 0, stream>>>(kappa, nu, scal);
    build_L_kernel<<<1024, 256, 0, stream>>>(diag, lower, L);
    llt_kernel<<<256, 128, 0, stream>>>(L, scatter);                 // 1024 waves
    stats_kernel<<<128, 256, 0, stream>>>(Xs, m, scal, x_mean, mu);
    sigma_kernel<<<65536, 256, 0, stream>>>(Xs, m, scal, scatter, x_mean, sigma);
    gj_inverse_kernel<<<CC, DD, 0, stream>>>(sigma, scal, biases);
    predict_kernel<<<2048, 128, 0, stream>>>(Xq, mu, sigma, biases, scal, out);
}